// DualSampling_87866440942276
// MI455X (gfx1250) — compile-verified
//
#include <hip/hip_runtime.h>
#include <hip/hip_bf16.h>

typedef float v2f __attribute__((ext_vector_type(2)));
typedef float v4f __attribute__((ext_vector_type(4)));
typedef float v8f __attribute__((ext_vector_type(8)));

#define NN     8192
#define DD     128
#define CHUNK  256
#define NCHUNK (NN / CHUNK)      // 32
#define ROWS   32                // two 16-row WMMA tiles per block
#define NWAVE  8                 // 256 threads
#define MASKW  (NN / 32)         // 256 words per row
#define SCP    (CHUNK + 4)       // pad: float4-aligned rows, banks shift by 4
#define NEGBIG (-3.0e38f)

// ---------------------------------------------------------------------------
// proj = user_emb @ W.T + b   (one wave per 16x16 tile, K=128 via f32 WMMA)
// ---------------------------------------------------------------------------
__global__ __launch_bounds__(32)
void ds_proj_kernel(const float* __restrict__ U, const float* __restrict__ W,
                    const float* __restrict__ bias, float* __restrict__ proj) {
    int tile = blockIdx.x;            // 512 * 8 tiles
    int it = tile >> 3, jt = tile & 7;
    int lane = threadIdx.x & 31;
    int half = lane >> 4, m16 = lane & 15;
    int ibase = it * 16, jbase = jt * 16;

    const float* aB = U + (size_t)(ibase + m16) * DD + half * 2;
    const float* bB = W + (size_t)(jbase + m16) * DD + half * 2;  // B[k][n] = W[n][k]

    v8f acc = {};
#pragma unroll
    for (int kk = 0; kk < 32; ++kk) {
        v2f a = *(const v2f*)(aB + kk * 4);
        v2f b = *(const v2f*)(bB + kk * 4);
        acc = __builtin_amdgcn_wmma_f32_16x16x4_f32(false, a, false, b,
                                                    (short)0, acc, false, false);
    }
    float bv = bias[jbase + m16];
#pragma unroll
    for (int v = 0; v < 8; ++v)
        proj[(size_t)(ibase + v + 8 * half) * DD + jbase + m16] = acc[v] + bv;
}

// ---------------------------------------------------------------------------
// mask bitmask: zero + diagonal, then scatter social edges
// ---------------------------------------------------------------------------
__global__ void ds_mask_init(unsigned* __restrict__ mask) {
    int w = blockIdx.x * blockDim.x + threadIdx.x;
    if (w >= NN * MASKW) return;
    int row  = w >> 8;               // MASKW == 256
    int word = w & 255;
    mask[w] = (word == (row >> 5)) ? (1u << (row & 31)) : 0u;
}

__global__ void ds_mask_scatter(const long long* __restrict__ edges,
                                unsigned* __restrict__ mask, int nE) {
    int e = blockIdx.x * blockDim.x + threadIdx.x;
    if (e >= nE) return;
    int r = (int)edges[e]      & (NN - 1);
    int c = (int)edges[nE + e] & (NN - 1);
    atomicOr(&mask[(size_t)r * MASKW + (c >> 5)], 1u << (c & 31));
}

// ---------------------------------------------------------------------------
// fused kernel, 32 rows per block:
//   A)  WMMA dot tiles (2 row-tiles per B load) -> raw dots into LDS
//   A2) coalesced float4 pass: *5 + gumbel (NT load), mask bits -> -inf,
//       NT-store float4 zero background into out
//   B)  exact streaming top-10 per row (wave argmax + carried list)
//   C)  scatter 1.0 at winners
// softmax is monotonic => rank by dot/TAU + gumbel directly.
// ---------------------------------------------------------------------------
__global__ __launch_bounds__(NWAVE * 32)
void ds_topk_kernel(const float* __restrict__ proj,
                    const float* __restrict__ gumbel,
                    const unsigned* __restrict__ mask,
                    float* __restrict__ out) {
    int ibase = blockIdx.x * ROWS;
    int lane = threadIdx.x & 31, wave = threadIdx.x >> 5;
    int half = lane >> 4, m16 = lane & 15;

    __shared__ float sc[ROWS][SCP];
    __shared__ float carVal[2][ROWS][10];
    __shared__ int   carIdx[2][ROWS][10];

    for (int i = threadIdx.x; i < ROWS * 10; i += NWAVE * 32) {
        carVal[0][i / 10][i % 10] = NEGBIG;
        carIdx[0][i / 10][i % 10] = 0;
    }
    __syncthreads();

    // hoist A fragments for both 16-row tiles (2 x 64 VGPRs)
    v2f a0[32], a1[32];
    {
        const float* aB0 = proj + (size_t)(ibase      + m16) * DD + half * 2;
        const float* aB1 = proj + (size_t)(ibase + 16 + m16) * DD + half * 2;
#pragma unroll
        for (int kk = 0; kk < 32; ++kk) {
            a0[kk] = *(const v2f*)(aB0 + kk * 4);
            a1[kk] = *(const v2f*)(aB1 + kk * 4);
        }
    }

    for (int c = 0; c < NCHUNK; ++c) {
        int cur = c & 1, nxt = cur ^ 1;
        int cbase = c * CHUNK;

        // ---- phase A: WMMA tiles, every B load feeds 2 accumulators ----
        for (int t = wave; t < CHUNK / 16; t += NWAVE) {
            int jbase = cbase + t * 16;
            const float* bB = proj + (size_t)(jbase + m16) * DD + half * 2;
            v8f acc0 = {}, acc1 = {};
#pragma unroll
            for (int kk = 0; kk < 32; ++kk) {
                v2f b = *(const v2f*)(bB + kk * 4);
                acc0 = __builtin_amdgcn_wmma_f32_16x16x4_f32(false, a0[kk], false, b,
                                                             (short)0, acc0, false, false);
                acc1 = __builtin_amdgcn_wmma_f32_16x16x4_f32(false, a1[kk], false, b,
                                                             (short)0, acc1, false, false);
            }
#pragma unroll
            for (int v = 0; v < 8; ++v) {
                sc[v + 8 * half][t * 16 + m16]      = acc0[v];
                sc[16 + v + 8 * half][t * 16 + m16] = acc1[v];
            }
        }
        __syncthreads();

        // ---- phase A2: coalesced score finalize + zero background ----
        for (int e = threadIdx.x; e < ROWS * (CHUNK / 4); e += NWAVE * 32) {
            int row = e / (CHUNK / 4);
            int c4  = e % (CHUNK / 4);
            int grow = ibase + row;
            int gcol = cbase + c4 * 4;
            v4f g = __builtin_nontemporal_load(
                        (const v4f*)(gumbel + (size_t)grow * NN + gcol));
            v4f* scp = (v4f*)&sc[row][c4 * 4];
            v4f s = *scp;
            s = s * 5.0f + g;                       // 1/TAU = 5
            unsigned mw = mask[(size_t)grow * MASKW + (gcol >> 5)];
            unsigned mb = (mw >> (gcol & 31)) & 0xFu;
            if (mb & 1u) s.x = NEGBIG;
            if (mb & 2u) s.y = NEGBIG;
            if (mb & 4u) s.z = NEGBIG;
            if (mb & 8u) s.w = NEGBIG;
            *scp = s;
            v4f z = {0.0f, 0.0f, 0.0f, 0.0f};
            __builtin_nontemporal_store(z, (v4f*)(out + (size_t)grow * NN + gcol));
        }
        __syncthreads();

        // ---- phase B: merge chunk with carried top-10 (exact streaming top-k) ----
        for (int r = wave; r < ROWS; r += NWAVE) {
#pragma unroll 1
            for (int it = 0; it < 10; ++it) {
                float best = -3.4e38f;
                int bpos = 1 << 20;
                for (int p = lane; p < CHUNK; p += 32) {
                    float v = sc[r][p];
                    if (v > best) { best = v; bpos = p; }
                }
                if (lane < 10) {
                    float v = carVal[cur][r][lane];
                    if (v > best) { best = v; bpos = CHUNK + lane; }
                }
                for (int off = 16; off > 0; off >>= 1) {
                    float ov = __shfl_xor(best, off);
                    int   op = __shfl_xor(bpos, off);
                    if (ov > best || (ov == best && op < bpos)) { best = ov; bpos = op; }
                }
                if (lane == 0) {
                    int gidx;
                    if (bpos < CHUNK) { gidx = cbase + bpos; sc[r][bpos] = NEGBIG; }
                    else { int s = bpos - CHUNK; gidx = carIdx[cur][r][s]; carVal[cur][r][s] = NEGBIG; }
                    carVal[nxt][r][it] = best;
                    carIdx[nxt][r][it] = gidx;
                }
                __builtin_amdgcn_wave_barrier();
            }
        }
        __syncthreads();
    }

    // ---- write the 10 winners per row (ordered after zeros by the barriers) ----
    int finalBuf = NCHUNK & 1;   // 32 chunks -> buffer 0
    for (int r = wave; r < ROWS; r += NWAVE) {
        if (lane < 10) {
            int idx = carIdx[finalBuf][r][lane];
            out[(size_t)(ibase + r) * NN + idx] = 1.0f;
        }
    }
}

// ---------------------------------------------------------------------------
extern "C" void kernel_launch(void* const* d_in, const int* in_sizes, int n_in,
                              void* d_out, int out_size, void* d_ws, size_t ws_size,
                              hipStream_t stream) {
    const float*     user_emb = (const float*)d_in[0];
    // d_in[1] = item_emb: unused by the reference computation
    const long long* edges    = (const long long*)d_in[2];   // int64 [2, 131072]
    const float*     W        = (const float*)d_in[3];
    const float*     bias     = (const float*)d_in[4];
    const float*     gumbel   = (const float*)d_in[5];
    float*           out      = (float*)d_out;

    int nE = in_sizes[2] / 2;

    // workspace layout: proj (4 MB) | mask bitmask (8 MB)
    float*    proj = (float*)d_ws;
    unsigned* mask = (unsigned*)((char*)d_ws + (size_t)NN * DD * sizeof(float));

    ds_proj_kernel<<<(NN / 16) * (DD / 16), 32, 0, stream>>>(user_emb, W, bias, proj);

    int maskWords = NN * MASKW;
    ds_mask_init<<<(maskWords + 255) / 256, 256, 0, stream>>>(mask);
    ds_mask_scatter<<<(nE + 255) / 256, 256, 0, stream>>>(edges, mask, nE);

    ds_topk_kernel<<<NN / ROWS, NWAVE * 32, 0, stream>>>(proj, gumbel, mask, out);
}